// DeconvDft2dLayer_46488726011904
// MI455X (gfx1250) — compile-verified
//
#include <hip/hip_runtime.h>

// DeconvDft2dLayer on MI455X (gfx1250).
// Reduction: the 8-tap filter lives only in row 0 of the padded grid, so the
// combined inverse spectrum gmf depends only on column frequency v:
//   gmf(v) = 1/|H(v)|^4   (real, even)
// => output = per-row 512-pt circular convolution with a real, EVEN kernel c:
//   Y[16384x512] = X[16384x512] @ C,  C[k][n] = c[(n-k) mod 512] symmetric.
// Implemented as an f32 WMMA GEMM (full fp32 accuracy, deterministic), with
// X tiles staged into padded LDS by the Tensor Data Mover (double-buffered,
// overlapped with compute) and B tiles generated from the 2KB vector c.

typedef __attribute__((ext_vector_type(2))) float v2f;
typedef __attribute__((ext_vector_type(8))) float v8f;
typedef __attribute__((ext_vector_type(4))) unsigned int v4u;
typedef __attribute__((ext_vector_type(8))) int v8i;
typedef __attribute__((ext_vector_type(4))) int v4i;

#define WDIM   512          // row length (and DFT size)
#define NROWS  16384        // 32 * 512 flattened rows
#define BLK_M  128
#define BLK_N  64
#define KC     64
#define XPAD   68           // LDS row stride (floats): 68 mod 64 = 4 -> conflict-free frags
#define BPAD   68

#define TWO_PI 6.28318530717958647692f

// ---------------------------------------------------------------------------
// Step 1: one block, 512 threads.
//   G(v) = 1 / |H(v)|^4,  H(v) = sum_{x=0..7} w[x] e^{-2pi i v x / 512}
//   c[m] = (1/512) * sum_v G(v) cos(2pi v m / 512)   (G even => c real, even)
// ---------------------------------------------------------------------------
__global__ __launch_bounds__(512) void build_circ_kernel(
    const float* __restrict__ w, float* __restrict__ cker) {
  __shared__ float Gs[WDIM];
  const int v = threadIdx.x;

  float re = 0.f, im = 0.f;
#pragma unroll
  for (int x = 0; x < 8; ++x) {
    const int ph = (v * x) & (WDIM - 1);          // angle reduced mod 2pi
    const float ang = (float)ph * (-TWO_PI / (float)WDIM);
    float s, c;
    __sincosf(ang, &s, &c);
    re += w[x] * c;
    im += w[x] * s;
  }
  const float p = re * re + im * im;              // |H(v)|^2
  Gs[v] = 1.0f / (p * p);                         // 1/|H(v)|^4
  __syncthreads();

  const int m = v;
  float acc = 0.f;
  for (int q = 0; q < WDIM; ++q) {
    const int ph = (q * m) & (WDIM - 1);
    const float ang = (float)ph * (TWO_PI / (float)WDIM);
    acc += Gs[q] * __cosf(ang);
  }
  cker[m] = acc * (1.0f / (float)WDIM);
}

// ---------------------------------------------------------------------------
// Tensor Data Mover: DMA one [BLK_M x KC] f32 sub-tile of the row-major
// [NROWS x 512] tensor into LDS with a 4-DWORD pad after every 64-DWORD row
// (=> LDS row stride 68 floats, matching Xs[][XPAD]).
// Descriptor per CDNA5 ISA section 8.3/8.4. Issued by one wave; EXEC ignored.
// ---------------------------------------------------------------------------
__device__ __forceinline__ void tdm_load_xtile(const float* gsrc,
                                               unsigned int lds_byte_off) {
  const unsigned long long ga = (unsigned long long)(uintptr_t)gsrc;
  v4u g0;
  g0.x = 1u;                                            // count=1, gather off
  g0.y = lds_byte_off;                                  // LDS dest (bytes)
  g0.z = (unsigned int)ga;                              // global_addr[31:0]
  g0.w = ((unsigned int)(ga >> 32) & 0x01FFFFFFu)       // global_addr[56:32]
         | 0x80000000u;                                 // type=2 ("image")
  v8i g1;
  g1[0] = (int)((2u << 16)      // data_size = 4 bytes
                | (1u << 20)    // pad_enable
                | (5u << 22)    // pad_interval: 64 DWORDs (one tile row)
                | (3u << 25));  // pad_amount: 4 DWORDs
  g1[1] = (int)((unsigned)WDIM << 16);   // tensor_dim0 = 512 (low 16 bits)
  g1[2] = (int)((unsigned)NROWS << 16);  // dim0 hi = 0 | tensor_dim1 = 16384
  g1[3] = (int)((unsigned)KC << 16);     // dim1 hi = 0 | tile_dim0 = 64
  g1[4] = BLK_M;                         // tile_dim1 = 128, tile_dim2 = 0
  g1[5] = WDIM;                          // tensor_dim0_stride = 512
  g1[6] = 0;
  g1[7] = 0;
  const v4i z4 = {0, 0, 0, 0};           // 2D tensor: groups 2/3 unused
  const v8i z8 = {0, 0, 0, 0, 0, 0, 0, 0};
  __builtin_amdgcn_tensor_load_to_lds(g0, g1, z4, z4, z8, 0);
}

// ---------------------------------------------------------------------------
// Step 2: Y = X @ C via v_wmma_f32_16x16x4_f32.
// Block: 256 threads = 8 waves in a 4(M) x 2(N) grid; wave tile 32x32
// (2 A-frags x 2 B-frags -> 4 wmma per k-step, 4x v8f accumulators).
// Double-buffered: TDM DMAs X chunk i+1 and the B tile for chunk i+1 is
// generated from cker while chunk i is consumed by the WMMAs.
// ---------------------------------------------------------------------------
__global__ __launch_bounds__(256) void circ_gemm(
    const float* __restrict__ X, const float* __restrict__ cker,
    float* __restrict__ Y) {
  __shared__ float Xs[2][BLK_M][XPAD];   // 2 x 34816 B
  __shared__ float Bs[2][BLK_N][BPAD];   // 2 x 17408 B

  const int n0   = blockIdx.x * BLK_N;
  const int m0   = blockIdx.y * BLK_M;
  const int tid  = threadIdx.x;
  const int lane = tid & 31;
  const int wave = tid >> 5;            // 0..7
  const int wm   = wave >> 1;           // 0..3 -> rows wm*32 .. wm*32+31
  const int wn   = wave & 1;            // 0..1 -> cols wn*32 .. wn*32+31
  const int l16  = lane & 15;
  const int hi   = lane >> 4;           // selects K pair {0,1} vs {2,3}

  v8f acc00 = {}, acc01 = {}, acc10 = {}, acc11 = {};

  // ---- prologue: stage chunk 0 ----
  if (wave == 0)
    tdm_load_xtile(X + (size_t)m0 * WDIM,
                   (unsigned int)(uintptr_t)&Xs[0][0][0]);
  {
#pragma unroll
    for (int it = 0; it < (BLK_N * KC) / 256; ++it) {   // 16 iters
      const int idx = tid + it * 256;
      const int n = idx >> 6;
      const int k = idx & (KC - 1);
      Bs[0][n][k] = cker[(k - (n0 + n)) & (WDIM - 1)];
    }
  }
  if (wave == 0) __builtin_amdgcn_s_wait_tensorcnt(0);
  __syncthreads();

  for (int ic = 0; ic < WDIM / KC; ++ic) {              // 8 chunks
    const int cur = ic & 1;
    const int nxt = cur ^ 1;
    const int kc_next = (ic + 1) * KC;

    // ---- overlap: DMA X chunk i+1, build B chunk i+1 ----
    if (ic < WDIM / KC - 1) {
      if (wave == 0)
        tdm_load_xtile(X + (size_t)m0 * WDIM + kc_next,
                       (unsigned int)(uintptr_t)&Xs[nxt][0][0]);
#pragma unroll
      for (int it = 0; it < (BLK_N * KC) / 256; ++it) {
        const int idx = tid + it * 256;
        const int n = idx >> 6;
        const int k = idx & (KC - 1);
        Bs[nxt][n][k] = cker[(kc_next + k - (n0 + n)) & (WDIM - 1)];
      }
    }

    // ---- compute chunk i: 16 k-steps x 4 WMMA ----
    const float (*Xc)[XPAD] = Xs[cur];
    const float (*Bc)[BPAD] = Bs[cur];
#pragma unroll
    for (int ks = 0; ks < KC; ks += 4) {
      // A frags: A[m][k], lane: m = l16, k = hi*2 + {0,1}
      const v2f a0 = *(const v2f*)&Xc[wm * 32 +  0 + l16][ks + hi * 2];
      const v2f a1 = *(const v2f*)&Xc[wm * 32 + 16 + l16][ks + hi * 2];
      // B frags: B[k][n] = C[k][n0+wn*32+j*16+l16] = Bt[n][k] (C symmetric)
      const v2f b0 = *(const v2f*)&Bc[wn * 32 +  0 + l16][ks + hi * 2];
      const v2f b1 = *(const v2f*)&Bc[wn * 32 + 16 + l16][ks + hi * 2];
      acc00 = __builtin_amdgcn_wmma_f32_16x16x4_f32(
          false, a0, false, b0, (short)0, acc00, false, false);
      acc01 = __builtin_amdgcn_wmma_f32_16x16x4_f32(
          false, a0, false, b1, (short)0, acc01, false, false);
      acc10 = __builtin_amdgcn_wmma_f32_16x16x4_f32(
          false, a1, false, b0, (short)0, acc10, false, false);
      acc11 = __builtin_amdgcn_wmma_f32_16x16x4_f32(
          false, a1, false, b1, (short)0, acc11, false, false);
    }

    if (wave == 0 && ic < WDIM / KC - 1)
      __builtin_amdgcn_s_wait_tensorcnt(0);
    __syncthreads();
  }

  // ---- epilogue: C/D layout -> VGPR r holds M = r (lanes 0-15), r+8 (16-31)
#pragma unroll
  for (int r = 0; r < 8; ++r) {
    const int gm0r = m0 + wm * 32 + r + 8 * hi;
    const int col  = n0 + wn * 32 + l16;
    Y[(size_t)(gm0r +  0) * WDIM + col +  0] = acc00[r];
    Y[(size_t)(gm0r +  0) * WDIM + col + 16] = acc01[r];
    Y[(size_t)(gm0r + 16) * WDIM + col +  0] = acc10[r];
    Y[(size_t)(gm0r + 16) * WDIM + col + 16] = acc11[r];
  }
}

extern "C" void kernel_launch(void* const* d_in, const int* in_sizes, int n_in,
                              void* d_out, int out_size, void* d_ws,
                              size_t ws_size, hipStream_t stream) {
  (void)in_sizes; (void)n_in; (void)out_size; (void)ws_size;
  const float* x = (const float*)d_in[0];   // [32,512,512,1] f32
  const float* w = (const float*)d_in[1];   // [1,8] f32
  float* out  = (float*)d_out;              // [32,512,512,1] f32
  float* cker = (float*)d_ws;               // 512 floats of scratch

  build_circ_kernel<<<1, 512, 0, stream>>>(w, cker);

  dim3 grid(WDIM / BLK_N, NROWS / BLK_M);   // (8, 128)
  circ_gemm<<<grid, 256, 0, stream>>>(x, cker, out);
}